// _DivergenceScore_42623255446045
// MI455X (gfx1250) — compile-verified
//
#include <hip/hip_runtime.h>

typedef __attribute__((ext_vector_type(2))) float v2f;
typedef __attribute__((ext_vector_type(8))) float v8f;

constexpr int Dfix   = 1024;   // feature dim
constexpr int Cfix   = 1000;   // classes
constexpr int DSLICE = 64;     // columns per segsum block
constexpr int TPB_B  = 512;    // 16 waves per segsum block
constexpr int BY     = 16;     // row-chunk blocks (65536/16 = 4096 rows/block)

// ---------------- zero accumulator region ----------------
__global__ void _dz_zero_kernel(float* __restrict__ ws, int n) {
    int i = blockIdx.x * blockDim.x + threadIdx.x;
    if (i < n) ws[i] = 0.0f;
}

// ---------------- per-class counts (LDS histogram) ----------------
__global__ __launch_bounds__(256) void _dz_count_kernel(
    const int* __restrict__ lbls, float* __restrict__ counts, int n) {
    __shared__ unsigned hist[Cfix];
    for (int i = threadIdx.x; i < Cfix; i += blockDim.x) hist[i] = 0u;
    __syncthreads();
    for (int i = blockIdx.x * blockDim.x + threadIdx.x; i < n;
         i += gridDim.x * blockDim.x) {
        atomicAdd(&hist[lbls[i]], 1u);
    }
    __syncthreads();
    for (int i = threadIdx.x; i < Cfix; i += blockDim.x) {
        unsigned h = hist[i];
        if (h) atomicAdd(&counts[i], (float)h);
    }
}

// ---------------- segment sum: feats[N,D] scattered by label ----------------
// Block (bx,by): columns [bx*64, bx*64+64), rows [by*4096, (by+1)*4096).
// One wave processes one row per iteration: lane L covers cols 2L, 2L+1.
// LDS accumulator: 1000 classes x 64 cols x f32 = 256 KB (<= 320 KB WGP LDS).
__global__ __launch_bounds__(TPB_B) void _dz_segsum_kernel(
    const float* __restrict__ feats, const int* __restrict__ lbls,
    float* __restrict__ sums, int nrows) {
    __shared__ float acc[Cfix * DSLICE];
    for (int i = threadIdx.x; i < Cfix * DSLICE; i += TPB_B) acc[i] = 0.0f;
    __syncthreads();

    const int lane     = threadIdx.x & 31;
    const int waveid   = threadIdx.x >> 5;           // 0..15
    const int colslice = blockIdx.x * DSLICE;
    const int rowsPerBlock = nrows / BY;
    const int row0   = blockIdx.y * rowsPerBlock;
    const int rowEnd = row0 + rowsPerBlock;

    for (int r = row0 + waveid; r < rowEnd; r += (TPB_B / 32)) {
        const int lbl = lbls[r];                      // wave-uniform
        const float* p = feats + (size_t)r * Dfix + colslice + lane * 2;
        v2f v = __builtin_nontemporal_load((const v2f*)p);  // streaming, no reuse
        float* a = &acc[lbl * DSLICE + lane * 2];     // 32 distinct banks/lane-op
        atomicAdd(a,     v.x);                        // ds_add_f32
        atomicAdd(a + 1, v.y);                        // ds_add_f32
    }
    __syncthreads();

    // Flush LDS tile -> global sums (L2-resident 4 MB accumulator)
    for (int i = threadIdx.x; i < Cfix * DSLICE; i += TPB_B) {
        float v = acc[i];
        if (v != 0.0f) {
            atomicAdd(&sums[(size_t)(i >> 6) * Dfix + colslice + (i & 63)], v);
        }
    }
}

// ---------------- per-class divergence via WMMA reduction ----------------
// One wave per class. B = all-ones => D[m,n] = sum_k A[m,k]; summing every
// C/D register over all 32 lanes gives 16 * (sum of all 64 A values),
// independent of the WMMA operand striping. 16 chained WMMAs cover D=1024.
__global__ __launch_bounds__(32) void _dz_classdiv_kernel(
    const float* __restrict__ sums,  const float* __restrict__ counts,
    const float* __restrict__ proto, const float* __restrict__ cov,
    float* __restrict__ partial, float* __restrict__ present) {
    const int c    = blockIdx.x;
    const int lane = threadIdx.x;
    const float cnt = counts[c];
    const float inv = 1.0f / fmaxf(cnt, 1.0f);   // means = sums / max(count,1)
    const size_t base = (size_t)c * Dfix;

    v8f accv = {};
    v2f bones; bones.x = 1.0f; bones.y = 1.0f;

#pragma unroll
    for (int it = 0; it < Dfix / 64; ++it) {
        const int d = it * 64 + lane * 2;
        v2f s = *(const v2f*)(sums  + base + d);
        v2f p = *(const v2f*)(proto + base + d);
        v2f q = *(const v2f*)(cov   + base + d);
        const float mx = s.x * inv - p.x;
        const float my = s.y * inv - p.y;
        v2f a;
        a.x = (mx * mx) / (q.x + 1e-6f);
        a.y = (my * my) / (q.y + 1e-6f);
        accv = __builtin_amdgcn_wmma_f32_16x16x4_f32(
            /*neg_a=*/false, a, /*neg_b=*/false, bones,
            /*c_mod=*/(short)0, accv, /*reuse_a=*/false, /*reuse_b=*/false);
    }

    float s8 = accv[0] + accv[1] + accv[2] + accv[3] +
               accv[4] + accv[5] + accv[6] + accv[7];
    for (int m = 16; m >= 1; m >>= 1) s8 += __shfl_xor(s8, m, 32);
    const float total = s8 * (1.0f / 16.0f);     // undo 16-column duplication

    if (lane == 0) {
        const bool pres = cnt > 0.0f;
        partial[c] = pres ? total : 0.0f;
        present[c] = pres ? 1.0f : 0.0f;
    }
}

// ---------------- deterministic final reduction ----------------
__global__ __launch_bounds__(1024) void _dz_finalize_kernel(
    const float* __restrict__ partial, const float* __restrict__ present,
    float* __restrict__ out) {
    __shared__ float s_t[1024];
    __shared__ float s_p[1024];
    const int t = threadIdx.x;
    float tv = 0.0f, pv = 0.0f;
    if (t < Cfix) { tv = partial[t]; pv = present[t]; }
    s_t[t] = tv; s_p[t] = pv;
    __syncthreads();
    for (int off = 512; off >= 1; off >>= 1) {
        if (t < off) { s_t[t] += s_t[t + off]; s_p[t] += s_p[t + off]; }
        __syncthreads();
    }
    if (t == 0) out[0] = s_t[0] / (s_p[0] * (float)Dfix);
}

extern "C" void kernel_launch(void* const* d_in, const int* in_sizes, int n_in,
                              void* d_out, int out_size, void* d_ws, size_t ws_size,
                              hipStream_t stream) {
    const float* feats = (const float*)d_in[0];
    const int*   lbls  = (const int*)d_in[1];
    const float* proto = (const float*)d_in[2];
    const float* cov   = (const float*)d_in[3];
    const int N = in_sizes[1];            // 65536

    float* ws      = (float*)d_ws;
    float* sums    = ws;                               // C*D floats
    float* counts  = ws + (size_t)Cfix * Dfix;         // C floats
    float* partial = counts + Cfix;                    // C floats
    float* present = partial + Cfix;                   // C floats

    const int nzero = Cfix * Dfix + Cfix;
    _dz_zero_kernel<<<(nzero + 255) / 256, 256, 0, stream>>>(ws, nzero);
    _dz_count_kernel<<<64, 256, 0, stream>>>(lbls, counts, N);
    _dz_segsum_kernel<<<dim3(Dfix / DSLICE, BY), TPB_B, 0, stream>>>(feats, lbls, sums, N);
    _dz_classdiv_kernel<<<Cfix, 32, 0, stream>>>(sums, counts, proto, cov, partial, present);
    _dz_finalize_kernel<<<1, 1024, 0, stream>>>(partial, present, (float*)d_out);
}